// GINConvSequentialRelu_45414984188293
// MI455X (gfx1250) — compile-verified
//
#include <hip/hip_runtime.h>

#define D 128
#define TILE_LD 132              // 16x128 tile padded to 132 floats/row (bank-conflict-free, keeps 8B alignment)
#define WAVES 4
#define ROWS_PER_BLOCK (WAVES * 16)

typedef __attribute__((ext_vector_type(2))) float v2f;
typedef __attribute__((ext_vector_type(8))) float v8f;
typedef unsigned int v4u __attribute__((ext_vector_type(4)));
typedef int v4i __attribute__((ext_vector_type(4)));
typedef int v8i __attribute__((ext_vector_type(8)));

#if defined(__has_builtin)
#if __has_builtin(__builtin_amdgcn_tensor_load_to_lds) && \
    __has_builtin(__builtin_amdgcn_s_wait_tensorcnt)
#define HAVE_TDM 1
#endif
#endif
#ifndef HAVE_TDM
#define HAVE_TDM 0
#endif

// ---------------------------------------------------------------------------
// Kernel 1: h = (1 + eps) * x   (h lives in d_out; also the scatter target)
// ---------------------------------------------------------------------------
__global__ void gin_init_kernel(const float* __restrict__ x,
                                const float* __restrict__ eps,
                                float* __restrict__ h, int total4) {
  const float s = 1.0f + eps[0];
  const float4* __restrict__ x4 = (const float4*)x;
  float4* __restrict__ h4 = (float4*)h;
  for (int i = blockIdx.x * blockDim.x + threadIdx.x; i < total4;
       i += gridDim.x * blockDim.x) {
    float4 v = x4[i];
    v.x *= s; v.y *= s; v.z *= s; v.w *= s;
    h4[i] = v;
  }
}

// ---------------------------------------------------------------------------
// Kernel 2: h[dst] += x[src] for every edge. 128 threads = one feature row,
// coalesced 512B gather + 128 coalesced global_atomic_add_f32 (no return).
// x (51MB) and h (51MB) are L2-resident (192MB L2) -> L2-atomic bound.
// Prefetch the next edge's source row to hide L2 gather latency.
// ---------------------------------------------------------------------------
__global__ void gin_scatter_kernel(const float* __restrict__ x,
                                   const int* __restrict__ src,
                                   const int* __restrict__ dst,
                                   float* __restrict__ h, int n_edges) {
  const int c = threadIdx.x;  // 0..127 feature index
  for (int e = blockIdx.x; e < n_edges; e += gridDim.x) {
    const int e2 = e + gridDim.x;
    if (e2 < n_edges) {
      const int s2 = src[e2];
      __builtin_prefetch(&x[(size_t)s2 * D + c], 0, 0);  // global_prefetch_b8
    }
    const int s = src[e];
    const int d = dst[e];
    atomicAdd(&h[(size_t)d * D + c], x[(size_t)s * D + c]);
  }
}

// ---------------------------------------------------------------------------
// TDM: DMA a 2D row-major f32 tensor (rows x cols, stride == cols) from
// global memory into LDS at byte offset lds_off. D# packed per CDNA5 ISA
// ch. 8.3/8.4 (count=1, type=2 "image", data_size=2 -> 4B, tile == tensor).
// Tracked with TENSORcnt.
// ---------------------------------------------------------------------------
#if HAVE_TDM
__device__ __forceinline__ void tdm_load_2d_f32(const void* gptr,
                                                unsigned lds_off,
                                                unsigned rows, unsigned cols) {
  const unsigned long long ga = (unsigned long long)(uintptr_t)gptr;
  v4u g0;
  g0.x = 1u;                                   // count=1, user-mode, no gather
  g0.y = lds_off;                              // lds_addr [63:32]
  g0.z = (unsigned)(ga & 0xFFFFFFFFu);         // global_addr[31:0]
  g0.w = (unsigned)((ga >> 32) & 0x1FFFFFFu)   // global_addr[56:32]
         | (2u << 30);                         // type=2 ("image")
  v8i g1;
  g1[0] = (int)(2u << 16);                     // wg_mask=0, data_size=2 (4B)
  g1[1] = (int)((cols & 0xFFFFu) << 16);       // tensor_dim0[15:0]
  g1[2] = (int)((cols >> 16) | ((rows & 0xFFFFu) << 16));  // dim0 hi | dim1 lo
  g1[3] = (int)((rows >> 16) | ((cols & 0xFFFFu) << 16));  // dim1 hi | tile_dim0
  g1[4] = (int)(rows & 0xFFFFu);               // tile_dim1 | tile_dim2=0
  g1[5] = (int)cols;                           // tensor_dim0_stride[31:0]
  g1[6] = 0;                                   // stride0 hi | stride1 lo
  g1[7] = 0;                                   // stride1 hi
  v4i z4 = {0, 0, 0, 0};
#if __clang_major__ >= 23
  v8i z8 = {0, 0, 0, 0, 0, 0, 0, 0};
  __builtin_amdgcn_tensor_load_to_lds(g0, g1, z4, z4, z8, 0);
#else
  __builtin_amdgcn_tensor_load_to_lds(g0, g1, z4, z4, 0);
#endif
}
#endif

// ---------------------------------------------------------------------------
// Kernel 3: in-place MLP on h:  h = relu(h@W1+b1) @ W2 + b2
// 4 waves/block; wave w owns rows [blockRow + 16w, +16). Weights DMA'd into
// LDS by the Tensor Data Mover (wave 0), overlapped with per-wave tile
// staging. Full-precision fp32 GEMM via V_WMMA_F32_16X16X4_F32.
//   A (16x4 f32): lane l holds M=l%16, K = k0 + 2*(l/16) + {0,1}   -> v2f
//   B (4x16 f32): lane l holds N=l%16, K = k0 + 2*(l/16) + {0,1}   -> v2f
//   C/D (16x16):  lane l col N=l%16, VGPR v row M = v + 8*(l/16)   -> v8f
// ---------------------------------------------------------------------------
__global__ void __launch_bounds__(128)
gin_mlp_kernel(float* __restrict__ h,   // in: (1+eps)x+agg, out: MLP result
               const float* __restrict__ W1, const float* __restrict__ b1,
               const float* __restrict__ W2, const float* __restrict__ b2,
               int n_nodes) {
  extern __shared__ float smem[];
  float* sW1 = smem;                 // 16384 floats @ LDS byte 0
  float* sW2 = smem + 16384;         // 16384 floats @ LDS byte 65536
  float* sB1 = smem + 32768;         // 128
  float* sB2 = smem + 32896;         // 128
  float* sTiles = smem + 33024;      // WAVES * 16 * TILE_LD

  const int tid = threadIdx.x;
  const int wave = tid >> 5;         // wave32
  const int lane = tid & 31;
  const int lhalf = lane >> 4;       // 0|1
  const int l16 = lane & 15;

#if HAVE_TDM
  // Wave 0 kicks off the weight DMA (global -> LDS, no VGPR traffic).
  if (wave == 0) {
    tdm_load_2d_f32(W1, 0u, D, D);
    tdm_load_2d_f32(W2, (unsigned)(16384 * sizeof(float)), D, D);
  }
#else
  {
    const float4* g1 = (const float4*)W1;
    const float4* g2 = (const float4*)W2;
    float4* s1 = (float4*)sW1;
    float4* s2 = (float4*)sW2;
    for (int i = tid; i < (D * D) / 4; i += blockDim.x) {
      s1[i] = g1[i];
      s2[i] = g2[i];
    }
  }
#endif
  if (tid < D) {
    sB1[tid] = b1[tid];
    sB2[tid] = b2[tid];
  }

  // Meanwhile every wave stages its own 16x128 row tile of h into LDS.
  const int rowBase = blockIdx.x * ROWS_PER_BLOCK + wave * 16;
  float* myT = sTiles + wave * (16 * TILE_LD);
#pragma unroll
  for (int r = 0; r < 16; ++r) {
    const int row = rowBase + r;
    float4 v = make_float4(0.f, 0.f, 0.f, 0.f);
    if (row < n_nodes) v = ((const float4*)(h + (size_t)row * D))[lane];
    *(float4*)(myT + r * TILE_LD + lane * 4) = v;
  }

#if HAVE_TDM
  if (wave == 0) __builtin_amdgcn_s_wait_tensorcnt(0);  // weights landed
#endif
  __syncthreads();

  v8f acc[8];

  // ---- GEMM1: acc = tile(16x128) @ W1(128x128) + b1, ReLU ----
#pragma unroll
  for (int nt = 0; nt < 8; ++nt) {
    const float bias = sB1[nt * 16 + l16];
    v8f t = {bias, bias, bias, bias, bias, bias, bias, bias};
    acc[nt] = t;
  }
  for (int k0 = 0; k0 < D; k0 += 4) {
    const int krow = k0 + 2 * lhalf;
    const v2f a = *(const v2f*)(myT + l16 * TILE_LD + krow);  // 8B ds_load
#pragma unroll
    for (int nt = 0; nt < 8; ++nt) {
      const int ncol = nt * 16 + l16;
      v2f b;
      b.x = sW1[krow * D + ncol];
      b.y = sW1[(krow + 1) * D + ncol];
      acc[nt] = __builtin_amdgcn_wmma_f32_16x16x4_f32(
          false, a, false, b, (short)0, acc[nt], false, false);
    }
  }
  // ReLU, write h1 back into the wave-private tile (DS ops in-order per wave).
#pragma unroll
  for (int nt = 0; nt < 8; ++nt) {
#pragma unroll
    for (int v = 0; v < 8; ++v) {
      const int m = v + 8 * lhalf;
      const int col = nt * 16 + l16;
      const float val = acc[nt][v];
      myT[m * TILE_LD + col] = val > 0.f ? val : 0.f;
    }
  }

  // ---- GEMM2: acc = h1(16x128) @ W2(128x128) + b2 ----
#pragma unroll
  for (int nt = 0; nt < 8; ++nt) {
    const float bias = sB2[nt * 16 + l16];
    v8f t = {bias, bias, bias, bias, bias, bias, bias, bias};
    acc[nt] = t;
  }
  for (int k0 = 0; k0 < D; k0 += 4) {
    const int krow = k0 + 2 * lhalf;
    const v2f a = *(const v2f*)(myT + l16 * TILE_LD + krow);
#pragma unroll
    for (int nt = 0; nt < 8; ++nt) {
      const int ncol = nt * 16 + l16;
      v2f b;
      b.x = sW2[krow * D + ncol];
      b.y = sW2[(krow + 1) * D + ncol];
      acc[nt] = __builtin_amdgcn_wmma_f32_16x16x4_f32(
          false, a, false, b, (short)0, acc[nt], false, false);
    }
  }

  // Store result (each block touches only its own rows -> in-place is safe).
#pragma unroll
  for (int nt = 0; nt < 8; ++nt) {
#pragma unroll
    for (int v = 0; v < 8; ++v) {
      const int m = v + 8 * lhalf;
      const int row = rowBase + m;
      const int col = nt * 16 + l16;
      if (row < n_nodes) h[(size_t)row * D + col] = acc[nt][v];
    }
  }
}

// ---------------------------------------------------------------------------
extern "C" void kernel_launch(void* const* d_in, const int* in_sizes, int n_in,
                              void* d_out, int out_size, void* d_ws, size_t ws_size,
                              hipStream_t stream) {
  const float* x   = (const float*)d_in[0];
  const int*   ei  = (const int*)d_in[1];
  const float* W1  = (const float*)d_in[2];
  const float* b1  = (const float*)d_in[3];
  const float* W2  = (const float*)d_in[4];
  const float* b2  = (const float*)d_in[5];
  const float* eps = (const float*)d_in[6];
  float* h = (float*)d_out;  // d_out doubles as the aggregation buffer

  const int n_nodes = in_sizes[0] / D;
  const int n_edges = in_sizes[1] / 2;
  const int* src = ei;            // edge_index[0]
  const int* dst = ei + n_edges;  // edge_index[1]

  const int total4 = (n_nodes * D) / 4;
  gin_init_kernel<<<2048, 256, 0, stream>>>(x, eps, h, total4);

  gin_scatter_kernel<<<8192, D, 0, stream>>>(x, src, dst, h, n_edges);

  const int grid = (n_nodes + ROWS_PER_BLOCK - 1) / ROWS_PER_BLOCK;
  const size_t smem =
      (size_t)(16384 + 16384 + 128 + 128 + WAVES * 16 * TILE_LD) * sizeof(float);
  (void)hipFuncSetAttribute((const void*)gin_mlp_kernel,
                            hipFuncAttributeMaxDynamicSharedMemorySize,
                            (int)smem);
  gin_mlp_kernel<<<grid, WAVES * 32, smem, stream>>>(h, W1, b1, W2, b2, n_nodes);
}